// QuantumGenerator_78245714198613
// MI455X (gfx1250) — compile-verified
//
#include <hip/hip_runtime.h>
#include <hip/hip_bf16.h>

typedef float v2f __attribute__((ext_vector_type(2)));
typedef float v8f __attribute__((ext_vector_type(8)));

#define TILES_PER_WAVE 8
#define WAVES_PER_BLOCK 8

// ---------------------------------------------------------------------------
// Kernel 1: build the constant 16x16 circuit matrices from the shared weights.
// U = (layer1 CNOTs)(layer1 Rots)(layer0 CNOTs)(layer0 Rots), complex 16x16.
// V[:,b] = U[:,b] * (-i)^popcount(b)   (folds product-state phases)
// G[k][j] = sum_i W[j][i] * sign_i(k)  (folds <Z_i> + Linear)
// Written to ws in WMMA B-operand layout:
//   B slice s, vgpr p, lane l  ->  B[4s + 2*(l>>4) + p][l & 15]
// ws[0..255]=B(Vr), ws[256..511]=B(Vi), ws[512..767]=B(G), ws[768..771]=bias
// ---------------------------------------------------------------------------
__global__ void qg_precompute(const float* __restrict__ qw,
                              const float* __restrict__ W,
                              const float* __restrict__ bias,
                              float* __restrict__ ws) {
    if (threadIdx.x != 0 || blockIdx.x != 0) return;

    float ur[16][16], ui[16][16];
    for (int i = 0; i < 16; ++i)
        for (int j = 0; j < 16; ++j) { ur[i][j] = (i == j) ? 1.0f : 0.0f; ui[i][j] = 0.0f; }

    for (int l = 0; l < 2; ++l) {
        // Rot(phi, theta, omega) on each wire (wire w -> bit (3-w) of flat idx)
        for (int w = 0; w < 4; ++w) {
            float phi = qw[l * 12 + w * 3 + 0];
            float th  = qw[l * 12 + w * 3 + 1];
            float om  = qw[l * 12 + w * 3 + 2];
            float ct = __cosf(0.5f * th), st = __sinf(0.5f * th);
            float ap = 0.5f * (phi + om), am = 0.5f * (phi - om);
            float cap = __cosf(ap), sap = __sinf(ap);
            float cam = __cosf(am), sam = __sinf(am);
            // g00 = e^{-i ap} ct ; g01 = -e^{+i am} st ; g10 = e^{-i am} st ; g11 = e^{+i ap} ct
            float g00r =  cap * ct, g00i = -sap * ct;
            float g01r = -cam * st, g01i = -sam * st;
            float g10r =  cam * st, g10i = -sam * st;
            float g11r =  cap * ct, g11i =  sap * ct;
            int mask = 1 << (3 - w);
            for (int i0 = 0; i0 < 16; ++i0) {
                if (i0 & mask) continue;
                int i1 = i0 | mask;
                for (int c = 0; c < 16; ++c) {
                    float ar = ur[i0][c], ai = ui[i0][c];
                    float br = ur[i1][c], bi = ui[i1][c];
                    ur[i0][c] = g00r * ar - g00i * ai + g01r * br - g01i * bi;
                    ui[i0][c] = g00r * ai + g00i * ar + g01r * bi + g01i * br;
                    ur[i1][c] = g10r * ar - g10i * ai + g11r * br - g11i * bi;
                    ui[i1][c] = g10r * ai + g10i * ar + g11r * bi + g11i * br;
                }
            }
        }
        // CNOT ring: control w, target (w+r)%4 ; left-mult = row swap where
        // control bit set (swap rows i <-> i^tmask).
        int r = (l % 3) + 1;
        for (int w = 0; w < 4; ++w) {
            int cm = 1 << (3 - w);
            int tm = 1 << (3 - ((w + r) & 3));
            for (int i = 0; i < 16; ++i) {
                if ((i & cm) && !(i & tm)) {
                    int i2 = i | tm;
                    for (int j = 0; j < 16; ++j) {
                        float tr = ur[i][j]; ur[i][j] = ur[i2][j]; ur[i2][j] = tr;
                        float ti = ui[i][j]; ui[i][j] = ui[i2][j]; ui[i2][j] = ti;
                    }
                }
            }
        }
    }

    // V = U * (-i)^popcount per column
    float vr[16][16], vi[16][16];
    for (int b = 0; b < 16; ++b) {
        int pc = __popc(b) & 3;
        for (int i = 0; i < 16; ++i) {
            float a = ur[i][b], c = ui[i][b];
            float rr, ii;
            if      (pc == 0) { rr =  a; ii =  c; }
            else if (pc == 1) { rr =  c; ii = -a; }   // * (-i)
            else if (pc == 2) { rr = -a; ii = -c; }   // * (-1)
            else              { rr = -c; ii =  a; }   // * (+i)
            vr[i][b] = rr; vi[i][b] = ii;
        }
    }

    // G[k][j] = sum_i W[j][i] * (bit(3-i) of k ? -1 : +1), j<4 else 0
    float G[16][16];
    for (int k = 0; k < 16; ++k)
        for (int j = 0; j < 16; ++j) {
            float g = 0.0f;
            if (j < 4)
                for (int i = 0; i < 4; ++i)
                    g += W[j * 4 + i] * (((k >> (3 - i)) & 1) ? -1.0f : 1.0f);
            G[k][j] = g;
        }

    // Emit B-operand layouts. For Pr = Prod @ Vr^T :  B[k][n] = Vr[n][k].
    for (int s = 0; s < 4; ++s)
        for (int p = 0; p < 2; ++p)
            for (int l2 = 0; l2 < 32; ++l2) {
                int n = l2 & 15;
                int k = 4 * s + 2 * (l2 >> 4) + p;
                ws[      (s * 2 + p) * 32 + l2] = vr[n][k];
                ws[256 + (s * 2 + p) * 32 + l2] = vi[n][k];
                ws[512 + (s * 2 + p) * 32 + l2] = G[k][n];
            }
    for (int j = 0; j < 4; ++j) ws[768 + j] = bias[j];
}

// ---------------------------------------------------------------------------
// Kernel 2: batch kernel. One wave = 16 batch elements per tile, 8 tiles/wave.
//   Pr = Prod @ Vr^T ; Pi = Prod @ Vi^T   (4+4 chained v_wmma_f32_16x16x4_f32)
//   Probs = Pr*Pr + Pi*Pi                 (elementwise, D layout)
//   LDS bounce: D layout -> A layout      (same-wave, s_wait_dscnt 0)
//   Out = Probs @ G                       (4 more WMMAs)
// ---------------------------------------------------------------------------
__global__ void __launch_bounds__(256) qg_main(const float* __restrict__ noise,
                                               const float* __restrict__ ws,
                                               float* __restrict__ out,
                                               int nTiles) {
    __shared__ float lds[WAVES_PER_BLOCK][256];

    const int lane = threadIdx.x & 31;
    const int wib  = threadIdx.x >> 5;
    const int waveGlobal = blockIdx.x * WAVES_PER_BLOCK + wib;
    const int m = lane & 15;   // batch element within tile (A-layout M / D-layout N)
    const int h = lane >> 4;   // lane half

    // Per-lane constant B operands (amortized over TILES_PER_WAVE tiles)
    v2f bvr[4], bvi[4], bg[4];
#pragma unroll
    for (int s = 0; s < 4; ++s) {
        bvr[s][0] = ws[      (s * 2 + 0) * 32 + lane];
        bvr[s][1] = ws[      (s * 2 + 1) * 32 + lane];
        bvi[s][0] = ws[256 + (s * 2 + 0) * 32 + lane];
        bvi[s][1] = ws[256 + (s * 2 + 1) * 32 + lane];
        bg[s][0]  = ws[512 + (s * 2 + 0) * 32 + lane];
        bg[s][1]  = ws[512 + (s * 2 + 1) * 32 + lane];
    }
    const float biasv = ws[768 + (lane & 3)];
    float* myLds = &lds[wib][0];

    for (int t = 0; t < TILES_PER_WAVE; ++t) {
        int tile = waveGlobal * TILES_PER_WAVE + t;
        if (tile >= nTiles) break;   // uniform across the wave

        int e = tile * 16 + m;
        const float4 x = ((const float4*)noise)[e];
        float c0 = __cosf(0.5f * x.x), s0 = __sinf(0.5f * x.x);
        float c1 = __cosf(0.5f * x.y), s1 = __sinf(0.5f * x.y);
        float c2 = __cosf(0.5f * x.z), s2 = __sinf(0.5f * x.z);
        float c3 = __cosf(0.5f * x.w), s3 = __sinf(0.5f * x.w);

        // Prod[m][b] in A-operand layout: lane needs b = 4s + 2h + {0,1}.
        v2f aP[4];
#pragma unroll
        for (int s = 0; s < 4; ++s) {
            int b0 = 4 * s + 2 * h;
            float p0 = ((b0 & 8) ? s0 : c0) * ((b0 & 4) ? s1 : c1) *
                       ((b0 & 2) ? s2 : c2) * ((b0 & 1) ? s3 : c3);
            int b1 = b0 + 1;
            float p1 = ((b1 & 8) ? s0 : c0) * ((b1 & 4) ? s1 : c1) *
                       ((b1 & 2) ? s2 : c2) * ((b1 & 1) ? s3 : c3);
            aP[s][0] = p0;
            aP[s][1] = p1;
        }

        v8f pr = {0, 0, 0, 0, 0, 0, 0, 0};
        v8f pi = {0, 0, 0, 0, 0, 0, 0, 0};
#pragma unroll
        for (int s = 0; s < 4; ++s) {
            pr = __builtin_amdgcn_wmma_f32_16x16x4_f32(false, aP[s], false, bvr[s],
                                                       (short)0, pr, false, false);
            pi = __builtin_amdgcn_wmma_f32_16x16x4_f32(false, aP[s], false, bvi[s],
                                                       (short)0, pi, false, false);
        }

        // probs in D layout: vgpr v, lane -> (M = v + 8h, N = m). Write to LDS [M][N].
#pragma unroll
        for (int v = 0; v < 8; ++v) {
            float q = pr[v] * pr[v] + pi[v] * pi[v];
            myLds[(v + 8 * h) * 16 + m] = q;
        }
        asm volatile("s_wait_dscnt 0" ::: "memory");   // same-wave LDS RAW

        // Read back in A-operand layout: A[m][k], k = 4s + 2h + {0,1}.
        v2f aQ[4];
#pragma unroll
        for (int s = 0; s < 4; ++s) {
            aQ[s][0] = myLds[m * 16 + 4 * s + 2 * h + 0];
            aQ[s][1] = myLds[m * 16 + 4 * s + 2 * h + 1];
        }

        v8f o = {0, 0, 0, 0, 0, 0, 0, 0};
#pragma unroll
        for (int s = 0; s < 4; ++s)
            o = __builtin_amdgcn_wmma_f32_16x16x4_f32(false, aQ[s], false, bg[s],
                                                      (short)0, o, false, false);

        // D layout: lane holds Out[v + 8h][j = m]; only j < 4 are real outputs.
        if (m < 4) {
            int j = m;
#pragma unroll
            for (int v = 0; v < 8; ++v) {
                int eo = tile * 16 + v + 8 * h;
                float val = o[v] + biasv;
                if (j == 0) val = __builtin_fabsf(val) + 0.001f;
                out[eo * 4 + j] = val;
            }
        }
    }
}

extern "C" void kernel_launch(void* const* d_in, const int* in_sizes, int n_in,
                              void* d_out, int out_size, void* d_ws, size_t ws_size,
                              hipStream_t stream) {
    const float* noise = (const float*)d_in[0];  // [B,4]
    const float* qw    = (const float*)d_in[1];  // [2,4,3]
    const float* W     = (const float*)d_in[2];  // [4,4]
    const float* b     = (const float*)d_in[3];  // [4]
    float* out = (float*)d_out;
    float* ws  = (float*)d_ws;

    int B = in_sizes[0] / 4;
    int nTiles = B / 16;   // B = 2^20 -> 65536 tiles

    qg_precompute<<<1, 1, 0, stream>>>(qw, W, b, ws);

    int wavesNeeded = (nTiles + TILES_PER_WAVE - 1) / TILES_PER_WAVE;
    int blocks = (wavesNeeded + WAVES_PER_BLOCK - 1) / WAVES_PER_BLOCK;
    qg_main<<<blocks, WAVES_PER_BLOCK * 32, 0, stream>>>(noise, ws, out, nTiles);
}